// RCBackwardEuler_5411658793505
// MI455X (gfx1250) — compile-verified
//
#include <hip/hip_runtime.h>
#include <stdint.h>

#define DEVINL __device__ __forceinline__

// ---------------- configuration ----------------
constexpr int THREADS = 128;          // 4 wave32s per block
constexpr int CHUNK   = 64;           // time steps per thread
constexpr int SE      = 8;            // steps per pipeline stage
constexpr int NST     = CHUNK / SE;   // 8 stages
constexpr int ROWW    = SE + 4;       // padded LDS row in words (48B, 16B aligned)
constexpr int ARRW    = THREADS * ROWW;   // 1536 words per array slab
constexpr int BUFW    = 5 * ARRW;         // 7680 words per buffer (5 arrays)
constexpr int SMEMW   = 2 * BUFW;         // 15360 words = 61,440 B (double buffer)
constexpr int TILE    = THREADS * CHUNK;  // 8192 elements per block

// ---------------- affine transform (x' = A x + c) ----------------
struct Aff { float a11, a12, a21, a22, c0, c1; };

DEVINL Aff aff_identity() { return {1.f, 0.f, 0.f, 1.f, 0.f, 0.f}; }

// apply f first, then g
DEVINL Aff aff_compose(const Aff& f, const Aff& g) {
    Aff r;
    r.a11 = g.a11 * f.a11 + g.a12 * f.a21;
    r.a12 = g.a11 * f.a12 + g.a12 * f.a22;
    r.a21 = g.a21 * f.a11 + g.a22 * f.a21;
    r.a22 = g.a21 * f.a12 + g.a22 * f.a22;
    r.c0  = g.a11 * f.c0  + g.a12 * f.c1 + g.c0;
    r.c1  = g.a21 * f.c0  + g.a22 * f.c1 + g.c1;
    return r;
}

DEVINL void aff_store(float* p, const Aff& a) {
    p[0] = a.a11; p[1] = a.a12; p[2] = a.a21; p[3] = a.a22; p[4] = a.c0; p[5] = a.c1;
}
DEVINL Aff aff_load(const float* p) {
    return {p[0], p[1], p[2], p[3], p[4], p[5]};
}

// ---------------- per-dispatch scalar constants ----------------
struct Consts {
    float M11, M12, M21, m12m21, inv_Rie;
    float k_to0, k_ir0, k_qi0;   // bu0 = k_to0*To + k_ir0*Irr + k_qi0*Qint
    float k_in,  k_ir1, k_qi1;   // bu1 = k_in*(To*invRia + Qah) + k_ir1*Irr + k_qi1*Qint
};

DEVINL Consts make_consts(float R_ie, float R_ea, float C_in, float C_en,
                          float a_sol_in, float a_sol_en, float a_int_in, float a_int_en) {
    const float dt = 1800.0f;
    const float GA = 0.76f * 41.3f;   // g * Az
    Consts K;
    float inv_Rie = 1.0f / R_ie;
    float inv_Rea = 1.0f / R_ea;
    float A11 = -(inv_Rea + inv_Rie) / C_en;
    float A12 = inv_Rie / C_en;
    float A21 = inv_Rie / C_in;
    K.M11 = 1.0f - dt * A11;
    K.M12 = -dt * A12;
    K.M21 = -dt * A21;
    K.m12m21 = K.M12 * K.M21;
    K.inv_Rie = inv_Rie;
    K.k_to0 = dt / (C_en * R_ea);
    K.k_ir0 = dt * (a_sol_en / C_en) * GA;
    K.k_qi0 = dt * (a_int_en / C_en);
    K.k_in  = dt / C_in;
    K.k_ir1 = dt * (a_sol_in / C_in) * GA;
    K.k_qi1 = dt * (a_int_in / C_in);
    return K;
}

// Per-step pieces (mirrors reference math)
DEVINL void step_core(const Consts& K, float To, float Irr, float Qint, float Qah, float Ria,
                      float& M22, float& invdet, float& b0, float& b1) {
    float invRia = 1.0f / fmaxf(Ria, 1e-6f);
    M22 = 1.0f + K.k_in * (invRia + K.inv_Rie);           // 1 - dt*A22
    float det = K.M11 * M22 - K.m12m21;
    invdet = 1.0f / det;
    b0 = K.k_to0 * To + K.k_ir0 * Irr + K.k_qi0 * Qint;
    b1 = K.k_in * (To * invRia + Qah) + K.k_ir1 * Irr + K.k_qi1 * Qint;
}

DEVINL Aff step_transform(const Consts& K, float To, float Irr, float Qint, float Qah, float Ria) {
    float M22, invdet, b0, b1;
    step_core(K, To, Irr, Qint, Qah, Ria, M22, invdet, b0, b1);
    Aff g;
    g.a11 =  M22   * invdet;
    g.a12 = -K.M12 * invdet;
    g.a21 = -K.M21 * invdet;
    g.a22 =  K.M11 * invdet;
    g.c0  = g.a11 * b0 + g.a12 * b1;
    g.c1  = g.a21 * b0 + g.a22 * b1;
    return g;
}

// ---------------- async staging through LDS (gfx1250 async-tensor path) ----------------
DEVINL unsigned lds_off_of(const void* p) {
    return (unsigned)(unsigned long long)(uintptr_t)p;    // low 32 bits of generic ptr = LDS offset
}

DEVINL void wait_async_le(int hold2) {
    if (hold2) asm volatile("s_wait_asynccnt 10" ::: "memory");
    else       asm volatile("s_wait_asynccnt 0"  ::: "memory");
}
DEVINL void wait_ds0() { asm volatile("s_wait_dscnt 0" ::: "memory"); }

// Issue one stage (SE steps per thread) of all 5 arrays: 10 async b128 ops per wave.
// Lane mapping gives coalesced 32B segments per lane-pair; LDS dest is the padded
// per-thread row so compute-side reads are contiguous float4s.
DEVINL void issue_stage(float* smem, int buf, int s, long long tileBase,
                        const float* p0, const float* p1, const float* p2,
                        const float* p3, const float* p4, long long N) {
    const float* arr[5] = {p0, p1, p2, p3, p4};
    int tid = threadIdx.x;
    int w = tid >> 5, l = tid & 31;
#pragma unroll
    for (int k = 0; k < 2; ++k) {
        int e  = k * 128 + l * 4;          // element id within wave-stage (0..255)
        int tw = e >> 3;                   // owning thread within wave
        int j  = e & 7;                    // 0 or 4
        int t  = w * 32 + tw;              // owning thread within block
        long long g = tileBase + (long long)t * CHUNK + (long long)s * SE + j;
        long long gmax = N - 4; if (gmax < 0) gmax = 0;
        if (g > gmax) g = gmax;
        unsigned ldsbase = lds_off_of(&smem[(buf * 5) * ARRW + t * ROWW + j]);
#pragma unroll
        for (int a = 0; a < 5; ++a) {
            unsigned loff = ldsbase + (unsigned)(a * ARRW * 4);
            unsigned long long gaddr = (unsigned long long)(uintptr_t)(arr[a] + g);
            asm volatile("global_load_async_to_lds_b128 %0, %1, off"
                         :: "v"(loff), "v"(gaddr) : "memory");
        }
    }
}

DEVINL void read_stage(const float* smem, int buf, int tid, float v[5][SE]) {
#pragma unroll
    for (int a = 0; a < 5; ++a) {
        const float* row = &smem[(buf * 5 + a) * ARRW + tid * ROWW];
        float4 x0 = *reinterpret_cast<const float4*>(row);
        float4 x1 = *reinterpret_cast<const float4*>(row + 4);
        v[a][0] = x0.x; v[a][1] = x0.y; v[a][2] = x0.z; v[a][3] = x0.w;
        v[a][4] = x1.x; v[a][5] = x1.y; v[a][6] = x1.z; v[a][7] = x1.w;
    }
}

// ---------------- Phase A: per-thread chunk transforms + block scan ----------------
__global__ __launch_bounds__(THREADS)
void rc_phaseA(const float* To, const float* Irr, const float* Qint, const float* Qah,
               const float* Ria,
               const float* pR_ie, const float* pR_ea, const float* pC_in, const float* pC_en,
               const float* pa_sol_in, const float* pa_sol_en,
               const float* pa_int_in, const float* pa_int_en,
               float* wsThread, float* wsBlock, long long N) {
    __shared__ __align__(16) float smem[SMEMW];
    Consts K = make_consts(*pR_ie, *pR_ea, *pC_in, *pC_en,
                           *pa_sol_in, *pa_sol_en, *pa_int_in, *pa_int_en);
    int tid = threadIdx.x;
    long long tileBase = (long long)blockIdx.x * TILE;
    long long myBase   = tileBase + (long long)tid * CHUNK;

    issue_stage(smem, 0, 0, tileBase, To, Irr, Qint, Qah, Ria, N);
    issue_stage(smem, 1, 1, tileBase, To, Irr, Qint, Qah, Ria, N);

    Aff F = aff_identity();
    for (int s = 0; s < NST; ++s) {
        wait_async_le(s < NST - 1);
        int buf = s & 1;
        float v[5][SE];
        read_stage(smem, buf, tid, v);
#pragma unroll
        for (int j = 0; j < SE; ++j) {
            long long g = myBase + (long long)s * SE + j;
            Aff gs = (g < N) ? step_transform(K, v[0][j], v[1][j], v[2][j], v[3][j], v[4][j])
                             : aff_identity();
            F = aff_compose(F, gs);
        }
        if (s + 2 < NST) { wait_ds0(); issue_stage(smem, buf, s + 2, tileBase, To, Irr, Qint, Qah, Ria, N); }
    }

    // Kogge-Stone inclusive scan over 128 thread transforms (reuse staging LDS)
    __syncthreads();
    float* scan = smem;               // needs 2*128*6 = 1536 floats
    int cur = 0;
    Aff val = F;
    aff_store(&scan[(cur * THREADS + tid) * 6], val);
    __syncthreads();
    for (int d = 1; d < THREADS; d <<= 1) {
        Aff nv = val;
        if (tid >= d) nv = aff_compose(aff_load(&scan[(cur * THREADS + (tid - d)) * 6]), val);
        aff_store(&scan[((cur ^ 1) * THREADS + tid) * 6], nv);
        __syncthreads();
        val = nv; cur ^= 1;
    }
    Aff excl = (tid == 0) ? aff_identity()
                          : aff_load(&scan[(cur * THREADS + (tid - 1)) * 6]);
    long long gt = (long long)blockIdx.x * THREADS + tid;
    aff_store(&wsThread[gt * 6], excl);
    if (tid == THREADS - 1) aff_store(&wsBlock[(long long)blockIdx.x * 6], val);
}

// ---------------- Phase B: scan block aggregates -> per-block start states ----------------
__global__ __launch_bounds__(1024)
void rc_phaseB(const float* wsBlock, float* wsState, const float* pTin0, int NB) {
    __shared__ float scan[2][1024 * 6];
    int tid = threadIdx.x;
    int q  = (NB + 1023) / 1024;
    int b0 = tid * q;
    int b1 = min(NB, b0 + q);

    Aff G = aff_identity();
    for (int b = b0; b < b1; ++b) G = aff_compose(G, aff_load(&wsBlock[(long long)b * 6]));

    int cur = 0;
    Aff val = G;
    aff_store(&scan[cur][tid * 6], val);
    __syncthreads();
    for (int d = 1; d < 1024; d <<= 1) {
        Aff nv = val;
        if (tid >= d) nv = aff_compose(aff_load(&scan[cur][(tid - d) * 6]), val);
        aff_store(&scan[cur ^ 1][tid * 6], nv);
        __syncthreads();
        val = nv; cur ^= 1;
    }
    Aff E = (tid == 0) ? aff_identity() : aff_load(&scan[cur][(tid - 1) * 6]);

    float T0 = *pTin0;
    float sx = E.a11 * T0 + E.a12 * T0 + E.c0;
    float sy = E.a21 * T0 + E.a22 * T0 + E.c1;
    for (int b = b0; b < b1; ++b) {
        wsState[(long long)b * 2 + 0] = sx;
        wsState[(long long)b * 2 + 1] = sy;
        Aff A = aff_load(&wsBlock[(long long)b * 6]);
        float nx = A.a11 * sx + A.a12 * sy + A.c0;
        float ny = A.a21 * sx + A.a22 * sy + A.c1;
        sx = nx; sy = ny;
    }
}

// ---------------- Phase C: replay with correct start state, emit Tin ----------------
__global__ __launch_bounds__(THREADS)
void rc_phaseC(const float* To, const float* Irr, const float* Qint, const float* Qah,
               const float* Ria,
               const float* pR_ie, const float* pR_ea, const float* pC_in, const float* pC_en,
               const float* pa_sol_in, const float* pa_sol_en,
               const float* pa_int_in, const float* pa_int_en,
               const float* wsThread, const float* wsState, float* out, long long N) {
    __shared__ __align__(16) float smem[SMEMW];
    Consts K = make_consts(*pR_ie, *pR_ea, *pC_in, *pC_en,
                           *pa_sol_in, *pa_sol_en, *pa_int_in, *pa_int_en);
    int tid = threadIdx.x;
    long long tileBase = (long long)blockIdx.x * TILE;
    long long myBase   = tileBase + (long long)tid * CHUNK;
    bool full = (tileBase + TILE) <= N;

    long long gt = (long long)blockIdx.x * THREADS + tid;
    Aff E = aff_load(&wsThread[gt * 6]);
    float sx = wsState[(long long)blockIdx.x * 2 + 0];
    float sy = wsState[(long long)blockIdx.x * 2 + 1];
    float x0 = E.a11 * sx + E.a12 * sy + E.c0;
    float x1 = E.a21 * sx + E.a22 * sy + E.c1;

    issue_stage(smem, 0, 0, tileBase, To, Irr, Qint, Qah, Ria, N);
    issue_stage(smem, 1, 1, tileBase, To, Irr, Qint, Qah, Ria, N);

    for (int s = 0; s < NST; ++s) {
        wait_async_le(s < NST - 1);
        int buf = s & 1;
        float v[5][SE];
        read_stage(smem, buf, tid, v);
        float ob[SE];
#pragma unroll
        for (int j = 0; j < SE; ++j) {
            float M22, invdet, b0, b1;
            step_core(K, v[0][j], v[1][j], v[2][j], v[3][j], v[4][j], M22, invdet, b0, b1);
            float r0 = x0 + b0, r1 = x1 + b1;
            float nx0 = ( M22   * r0 - K.M12 * r1) * invdet;
            float nx1 = (-K.M21 * r0 + K.M11 * r1) * invdet;
            x0 = nx0; x1 = nx1;
            ob[j] = nx1;
        }
        long long obase = myBase + (long long)s * SE;
        if (full) {
            *reinterpret_cast<float4*>(&out[obase])     = make_float4(ob[0], ob[1], ob[2], ob[3]);
            *reinterpret_cast<float4*>(&out[obase + 4]) = make_float4(ob[4], ob[5], ob[6], ob[7]);
        } else {
#pragma unroll
            for (int j = 0; j < SE; ++j)
                if (obase + j < N) out[obase + j] = ob[j];
        }
        if (s + 2 < NST) { wait_ds0(); issue_stage(smem, buf, s + 2, tileBase, To, Irr, Qint, Qah, Ria, N); }
    }
}

// ---------------- host-side launch ----------------
extern "C" void kernel_launch(void* const* d_in, const int* in_sizes, int n_in,
                              void* d_out, int out_size, void* d_ws, size_t ws_size,
                              hipStream_t stream) {
    // setup_inputs() order:
    const float* pR_ie     = (const float*)d_in[0];
    const float* pR_ea     = (const float*)d_in[1];
    const float* pC_in     = (const float*)d_in[2];
    const float* pC_en     = (const float*)d_in[3];
    const float* pa_sol_in = (const float*)d_in[4];
    const float* pa_sol_en = (const float*)d_in[5];
    const float* pa_int_in = (const float*)d_in[6];
    const float* pa_int_en = (const float*)d_in[7];
    const float* pTin0     = (const float*)d_in[8];
    const float* To        = (const float*)d_in[9];
    const float* Irr       = (const float*)d_in[10];
    const float* Qint      = (const float*)d_in[11];
    const float* Qah       = (const float*)d_in[12];
    const float* Ria       = (const float*)d_in[13];

    long long N = (long long)in_sizes[9];
    int NB = (int)((N + TILE - 1) / TILE);               // 512 for N = 4,194,304

    float* ws = (float*)d_ws;
    long long NT = (long long)NB * THREADS;
    float* wsThread = ws;                                // NT * 6 floats
    float* wsBlock  = ws + NT * 6;                       // NB * 6 floats
    float* wsState  = wsBlock + (long long)NB * 6;       // NB * 2 floats

    rc_phaseA<<<NB, THREADS, 0, stream>>>(To, Irr, Qint, Qah, Ria,
                                          pR_ie, pR_ea, pC_in, pC_en,
                                          pa_sol_in, pa_sol_en, pa_int_in, pa_int_en,
                                          wsThread, wsBlock, N);
    rc_phaseB<<<1, 1024, 0, stream>>>(wsBlock, wsState, pTin0, NB);
    rc_phaseC<<<NB, THREADS, 0, stream>>>(To, Irr, Qint, Qah, Ria,
                                          pR_ie, pR_ea, pC_in, pC_en,
                                          pa_sol_in, pa_sol_en, pa_int_in, pa_int_en,
                                          wsThread, wsState, (float*)d_out, N);
}